// MultiHeadAtt_39865886442006
// MI455X (gfx1250) — compile-verified
//
#include <hip/hip_runtime.h>
#include <math.h>

typedef __attribute__((ext_vector_type(2))) float v2f;
typedef __attribute__((ext_vector_type(8))) float v8f;

#define N_NODES  131072
#define H_HEADS  4
#define D_DIM    128
#define HD       512
#define B_GRAPHS 64
#define NPG      2048
#define E_EDGES  2097152
#define K_PER    1024
#define K_TOTAL  65536

// ---------------- K0: init deg=1 (self loop), newid=-1 ----------------
__global__ void k_init(int* __restrict__ deg, int* __restrict__ newid) {
    int n = blockIdx.x * 256 + threadIdx.x;
    deg[n] = 1;
    newid[n] = -1;
}

// ---------------- K1: in-degree over edges ----------------
__global__ void k_degree(const int* __restrict__ dst, int* __restrict__ deg) {
    int e = blockIdx.x * 256 + threadIdx.x;
    atomicAdd(&deg[dst[e]], 1);
}

// ---------------- K2: h_lin via WMMA f32 16x16x4 ----------------
// Each wave: 16-node tile. Per head: S = X(16x128) * W(128x2), W = [att_w_h | gcn_w_h],
// then hl += S[:,0]*S[:,1].  B operand is pre-striped in LDS in the exact per-lane
// WMMA 4x16 layout so the hot loop is branch-free (no exec-mask manipulation).
__global__ void k_hlin_wmma(const float* __restrict__ sub_xs,
                            const float* __restrict__ att_w,
                            const float* __restrict__ gcn_w,
                            float* __restrict__ hlin) {
    // [h][kstep][lane] -> B pair for that lane at that k-step (32 KB)
    __shared__ __align__(16) v2f bLDS[H_HEADS * 32 * 32];
    const int tid = threadIdx.x;
    for (int e = tid; e < H_HEADS * 32 * 32; e += 256) {
        int lane = e & 31;
        int ks   = (e >> 5) & 31;
        int h    = e >> 10;
        int kb   = (lane < 16) ? 0 : 2;      // K-pair owned by this half-wave
        int col  = lane & 15;                // N (output column)
        int k    = ks * 4 + kb;
        v2f v = {0.f, 0.f};
        if (col == 0)      { v.x = att_w[h * D_DIM + k]; v.y = att_w[h * D_DIM + k + 1]; }
        else if (col == 1) { v.x = gcn_w[h * D_DIM + k]; v.y = gcn_w[h * D_DIM + k + 1]; }
        bLDS[e] = v;
    }
    __syncthreads();

    const int wave = tid >> 5;
    const int lane = tid & 31;
    const int tile = blockIdx.x * 8 + wave;   // 8192 tiles total, exact
    const int n0   = tile * 16;
    const int M    = lane & 15;
    const int kb   = (lane < 16) ? 0 : 2;

    float hacc[8];
    #pragma unroll
    for (int j = 0; j < 8; ++j) hacc[j] = 0.f;

    #pragma unroll
    for (int h = 0; h < H_HEADS; ++h) {
        const float* xrow = sub_xs + ((size_t)h * N_NODES + (size_t)(n0 + M)) * D_DIM;
        const v2f*   bL   = bLDS + h * 32 * 32 + lane;

        v8f c = {0.f, 0.f, 0.f, 0.f, 0.f, 0.f, 0.f, 0.f};
        #pragma unroll
        for (int ks = 0; ks < 32; ++ks) {
            v2f a = *(const v2f*)(xrow + ks * 4 + kb);   // global_load_b64
            v2f b = bL[ks * 32];                          // ds_load_b64, imm offset
            c = __builtin_amdgcn_wmma_f32_16x16x4_f32(
                    false, a, false, b, (short)0, c, false, false);
        }
        // s1 in lanes 0/16, s2 in lanes 1/17 -> product via one xor-shuffle each.
        #pragma unroll
        for (int j = 0; j < 8; ++j) {
            float other = __shfl_xor(c[j], 1, 32);
            hacc[j] += c[j] * other;   // lane 0: node j ; lane 16: node j+8
        }
    }

    // redistribute: lane t (0..15) takes node t
    float hl = 0.f;
    #pragma unroll
    for (int j = 0; j < 8; ++j) {
        float b0  = __shfl(hacc[j], 0, 32);
        float b16 = __shfl(hacc[j], 16, 32);
        if (lane == j)     hl = b0;
        if (lane == j + 8) hl = b16;
    }
    if (lane < 16) hlin[n0 + lane] = hl;
}

// ---------------- K3: dinv + self-loop contribution ----------------
__global__ void k_dinv_self(const int* __restrict__ deg,
                            const float* __restrict__ hlin,
                            float* __restrict__ dinv,
                            float* __restrict__ acc) {
    int n = blockIdx.x * 256 + threadIdx.x;
    float di = rsqrtf((float)deg[n]);   // deg >= 1 always (self loop)
    dinv[n] = di;
    acc[n]  = di * di * hlin[n];
}

// ---------------- K4: normalized edge scatter-add ----------------
__global__ void k_scatter(const int* __restrict__ src, const int* __restrict__ dst,
                          const float* __restrict__ dinv,
                          const float* __restrict__ hlin,
                          float* __restrict__ acc) {
    int e = blockIdx.x * 256 + threadIdx.x;
    int s = src[e], d = dst[e];
    atomicAdd(&acc[d], dinv[s] * dinv[d] * hlin[s]);
}

// ---------------- K5: per-graph bitonic top-k (full sort of 2048) ----------------
__device__ __forceinline__ unsigned int fkey(float f) {
    unsigned int u = __float_as_uint(f);
    return (u & 0x80000000u) ? ~u : (u | 0x80000000u);
}

__global__ void k_topk(const float* __restrict__ acc,
                       int* __restrict__ perm_int, int* __restrict__ newid,
                       float* __restrict__ out_batch, float* __restrict__ out_perm) {
    __shared__ unsigned long long keys[NPG];   // 16 KB
    const int b = blockIdx.x;
    const int t = threadIdx.x;                 // 1024 threads
    const float* sc = acc + b * NPG;

    #pragma unroll
    for (int r = 0; r < 2; ++r) {
        int i = t + r * 1024;
        // ascending sort of (~fkey(score), idx) => descending score, ties lower idx first
        keys[i] = ((unsigned long long)(~fkey(sc[i])) << 32) | (unsigned int)i;
    }
    __syncthreads();

    for (unsigned int k = 2; k <= NPG; k <<= 1) {
        for (unsigned int j = k >> 1; j > 0; j >>= 1) {
            #pragma unroll
            for (int r = 0; r < 2; ++r) {
                unsigned int i   = t + r * 1024;
                unsigned int ixj = i ^ j;
                if (ixj > i) {
                    unsigned long long a = keys[i], bb = keys[ixj];
                    bool asc = ((i & k) == 0);
                    if (asc ? (a > bb) : (a < bb)) { keys[i] = bb; keys[ixj] = a; }
                }
            }
            __syncthreads();
        }
    }

    if (t < K_PER) {
        int idx  = (int)(keys[t] & 0xFFFFFFFFu);
        int node = b * NPG + idx;
        int pos  = b * K_PER + t;
        perm_int[pos]  = node;
        newid[node]    = pos;
        out_batch[pos] = (float)b;
        out_perm[pos]  = (float)node;
    }
}

// ---------------- K6: gather + tanh gate ----------------
__global__ void k_gate(const float* __restrict__ x,
                       const float* __restrict__ acc,
                       const float* __restrict__ gcn_b,
                       const int* __restrict__ perm_int,
                       float* __restrict__ out_x) {
    int gid = blockIdx.x * 256 + threadIdx.x;      // K_TOTAL * 128 threads
    int row = gid >> 7;
    int col = (gid & 127) * 4;
    int p   = perm_int[row];
    float g = tanhf(acc[p] + gcn_b[0]);
    const float4 v = *(const float4*)(x + (size_t)p * HD + col);
    float4 o;
    o.x = v.x * g; o.y = v.y * g; o.z = v.z * g; o.w = v.w * g;
    *(float4*)(out_x + (size_t)row * HD + col) = o;
}

// ---------------- K7: edge relabel + mask ----------------
__global__ void k_edges(const int* __restrict__ src, const int* __restrict__ dst,
                        const int* __restrict__ newid,
                        float* __restrict__ out_e, float* __restrict__ out_mask) {
    int e = blockIdx.x * 256 + threadIdx.x;
    int s = newid[src[e]];
    int d = newid[dst[e]];
    bool m = (s >= 0) && (d >= 0);
    out_e[e]           = m ? (float)s : -1.0f;
    out_e[E_EDGES + e] = m ? (float)d : -1.0f;
    out_mask[e]        = m ? 1.0f : 0.0f;
}

extern "C" void kernel_launch(void* const* d_in, const int* in_sizes, int n_in,
                              void* d_out, int out_size, void* d_ws, size_t ws_size,
                              hipStream_t stream) {
    const float* x      = (const float*)d_in[0];
    const float* sub_xs = (const float*)d_in[1];
    const float* att_w  = (const float*)d_in[2];
    const float* gcn_w  = (const float*)d_in[3];
    const float* gcn_b  = (const float*)d_in[4];
    const int*   ei     = (const int*)d_in[5];
    const int* src = ei;
    const int* dst = ei + E_EDGES;

    // workspace layout
    char* base = (char*)d_ws;
    int*   deg      = (int*)(base + 0);
    int*   newid    = (int*)(base + 524288);
    int*   perm_int = (int*)(base + 1048576);
    float* hlin     = (float*)(base + 1310720);
    float* dinv     = (float*)(base + 1835008);
    float* acc      = (float*)(base + 2359296);

    // output layout (all float32, concatenated in return order)
    float* out_x     = (float*)d_out;
    float* out_e     = out_x + (size_t)K_TOTAL * HD;       // 33554432
    float* out_mask  = out_e + 2 * (size_t)E_EDGES;
    float* out_batch = out_mask + (size_t)E_EDGES;
    float* out_perm  = out_batch + (size_t)K_TOTAL;

    k_init     <<<N_NODES / 256, 256, 0, stream>>>(deg, newid);
    k_degree   <<<E_EDGES / 256, 256, 0, stream>>>(dst, deg);
    k_hlin_wmma<<<N_NODES / (16 * 8), 256, 0, stream>>>(sub_xs, att_w, gcn_w, hlin);
    k_dinv_self<<<N_NODES / 256, 256, 0, stream>>>(deg, hlin, dinv, acc);
    k_scatter  <<<E_EDGES / 256, 256, 0, stream>>>(src, dst, dinv, hlin, acc);
    k_topk     <<<B_GRAPHS, 1024, 0, stream>>>(acc, perm_int, newid, out_batch, out_perm);
    k_gate     <<<(K_TOTAL * 128) / 256, 256, 0, stream>>>(x, acc, gcn_b, perm_int, out_x);
    k_edges    <<<E_EDGES / 256, 256, 0, stream>>>(src, dst, newid, out_e, out_mask);
}